// Match_62577673502813
// MI455X (gfx1250) — compile-verified
//
#include <hip/hip_runtime.h>
#include <math.h>

// ---------------------------------------------------------------------------
// Shapes
// ---------------------------------------------------------------------------
#define N_NODES 512
#define N_CLS   151
#define E_EDGES 50000
#define E_CLS   51
#define DIM     1024
#define NMS_THRESH 0.3f

typedef float v2f __attribute__((ext_vector_type(2)));
typedef float v4f __attribute__((ext_vector_type(4)));
typedef float v8f __attribute__((ext_vector_type(8)));

// ---------------------------------------------------------------------------
// Pack a K x N (strided) fp32 matrix into WMMA f32 16x16x4 B-fragment-major
// layout, with K-steps grouped in PAIRS so the GEMM can fetch two steps with
// one b128 load:
//   dst[(((pair*Ntiles + ntile)*32) + lane)*4 + e],  e = (s&1)*2 + j
//   s = 2*pair + (e>>1), fragment element j in {0,1}:
//     lane<16 : B[4s + j       ][ntile*16 + lane]
//     lane>=16: B[4s + 2 + j   ][ntile*16 + lane-16]
// Out-of-range (k>=K or n>=Nact) elements are zero.
// ---------------------------------------------------------------------------
__global__ void pack_b_kernel(const float* __restrict__ src,
                              int K, int strideK, int strideN, int Nact,
                              int Kpairs, int Ntiles,
                              float* __restrict__ dst) {
    int tid = blockIdx.x * blockDim.x + threadIdx.x;
    int total = Kpairs * Ntiles * 32;
    if (tid >= total) return;
    int lane = tid & 31;
    int rest = tid >> 5;
    int t = rest % Ntiles;
    int p = rest / Ntiles;
    int n = t * 16 + (lane & 15);
    int klane = (lane >> 4) << 1;           // 0 or 2
    float v[4];
#pragma unroll
    for (int e = 0; e < 4; ++e) {
        int s = 2 * p + (e >> 1);
        int j = e & 1;
        int k = s * 4 + klane + j;
        float val = 0.f;
        if (n < Nact && k < K)
            val = src[(size_t)k * strideK + (size_t)n * strideN];
        v[e] = val;
    }
    float* d = dst + (size_t)tid * 4;
    d[0] = v[0]; d[1] = v[1]; d[2] = v[2]; d[3] = v[3];
}

// ---------------------------------------------------------------------------
// WMMA fp32 GEMM: Out(M x Nact) = A(M x K, row-major lda) * B(packed frags)
// Each wave handles one 16-row M-tile and NTG consecutive 16-col N-tiles.
// blockDim = 256 (8 waves). grid.x covers M-tiles/8, grid.y covers N-groups.
// K loop runs over PAIRS of K-steps (K padded to multiple of 8 by packing).
// ---------------------------------------------------------------------------
template <int NTG>
__global__ __launch_bounds__(256) void wmma_gemm_kernel(
    const float* __restrict__ A, int lda,
    const float* __restrict__ Bp, int ntilesTotal,
    float* __restrict__ Out, int ldo,
    int Mtiles, int Nact, int Kpairs) {
    int wave = threadIdx.x >> 5;
    int lane = threadIdx.x & 31;
    int mtile = blockIdx.x * 8 + wave;
    if (mtile >= Mtiles) return;            // uniform per-wave: EXEC stays all-1s
    int tbase = blockIdx.y * NTG;

    v8f acc[NTG];
#pragma unroll
    for (int t = 0; t < NTG; ++t) acc[t] = v8f{0.f, 0.f, 0.f, 0.f, 0.f, 0.f, 0.f, 0.f};

    // A fragment addressing (16x4 fp32 layout):
    //   lanes 0-15 : row = lane,    k..k+1
    //   lanes 16-31: row = lane-16, k+2..k+3
    int arow  = mtile * 16 + (lane & 15);
    int akoff = (lane >> 4) << 1;
    const float* aptr = A + (size_t)arow * lda + akoff;

    for (int p = 0; p < Kpairs; ++p) {
        v2f a0 = *(const v2f*)(aptr + (size_t)p * 8);
        v2f a1 = *(const v2f*)(aptr + (size_t)p * 8 + 4);
        const float* bp = Bp + ((size_t)(p * ntilesTotal + tbase) * 32 + lane) * 4;
#pragma unroll
        for (int t = 0; t < NTG; ++t) {
            v4f b2 = *(const v4f*)(bp + t * 128);
            v2f b0 = __builtin_shufflevector(b2, b2, 0, 1);
            v2f b1 = __builtin_shufflevector(b2, b2, 2, 3);
            acc[t] = __builtin_amdgcn_wmma_f32_16x16x4_f32(
                false, a0, false, b0, (short)0, acc[t], false, false);
            acc[t] = __builtin_amdgcn_wmma_f32_16x16x4_f32(
                false, a1, false, b1, (short)0, acc[t], false, false);
        }
    }

    // C/D layout: VGPR r -> lanes 0-15: (M=r, N=lane); lanes 16-31: (M=8+r, N=lane-16)
    int col0   = lane & 15;
    int rowadd = (lane >> 4) * 8;
#pragma unroll
    for (int t = 0; t < NTG; ++t) {
        int col = (tbase + t) * 16 + col0;
        if (col < Nact) {
#pragma unroll
            for (int r = 0; r < 8; ++r) {
                int row = mtile * 16 + rowadd + r;
                Out[(size_t)row * ldo + col] = acc[t][r];
            }
        }
    }
}

// ---------------------------------------------------------------------------
// Zero fill
// ---------------------------------------------------------------------------
__global__ void zero_kernel(float* __restrict__ p, int n) {
    int i = blockIdx.x * blockDim.x + threadIdx.x;
    if (i < n) p[i] = 0.f;
}

// ---------------------------------------------------------------------------
// Softmax over 151 classes per node; one wave32 per row.
// Writes att (ld=152) with col 0 and pad col 151 forced to zero
// (col 0 is masked by nms_mask anyway; full sum still includes col 0).
// ---------------------------------------------------------------------------
__global__ __launch_bounds__(256) void softmax_kernel(const float* __restrict__ raw,
                                                      float* __restrict__ att) {
    int wave = threadIdx.x >> 5;
    int lane = threadIdx.x & 31;
    int row = blockIdx.x * 8 + wave;
    if (row >= N_NODES) return;
    const float* r = raw + (size_t)row * N_CLS;
    float mx = -INFINITY;
    for (int c = lane; c < N_CLS; c += 32) mx = fmaxf(mx, r[c]);
    for (int o = 16; o; o >>= 1) mx = fmaxf(mx, __shfl_xor(mx, o, 32));
    float sum = 0.f;
    for (int c = lane; c < N_CLS; c += 32) sum += expf(r[c] - mx);
    for (int o = 16; o; o >>= 1) sum += __shfl_xor(sum, o, 32);
    float inv = 1.0f / sum;
    float* a = att + (size_t)row * 152;
    for (int c = lane; c < 152; c += 32) {
        float v = (c == 0 || c >= N_CLS) ? 0.f : expf(r[c] - mx) * inv;
        a[c] = v;
    }
}

// ---------------------------------------------------------------------------
// Greedy NMS, one workgroup (256 threads) per class c in [1, 150].
// Bitonic sort of (score, idx) in LDS: score descending, idx ascending
// (matches stable argsort(-scores)). Then sequential greedy suppression.
// Suppressed entries zero att[node][c] in place.
// ---------------------------------------------------------------------------
__global__ __launch_bounds__(256) void nms_kernel(const float* __restrict__ boxes,
                                                  float* __restrict__ att) {
    const int NB = N_NODES;
    int c = blockIdx.x + 1;     // classes 1..150
    int tid = threadIdx.x;
    __shared__ float ss[NB];
    __shared__ int   si[NB];
    __shared__ float bx1[NB], by1[NB], bx2[NB], by2[NB], ar[NB];
    __shared__ int   supp[NB];

    for (int n = tid; n < NB; n += 256) {
        ss[n] = att[(size_t)n * 152 + c];
        si[n] = n;
    }
    __syncthreads();

    // bitonic sort: comes-first relation L(a,b) = a.s > b.s || (a.s==b.s && a.i < b.i)
    for (int k = 2; k <= NB; k <<= 1) {
        for (int j = k >> 1; j > 0; j >>= 1) {
            for (int i = tid; i < NB; i += 256) {
                int ixj = i ^ j;
                if (ixj > i) {
                    float s1 = ss[i], s2 = ss[ixj];
                    int   i1 = si[i], i2 = si[ixj];
                    bool partnerFirst = (s2 > s1) || (s2 == s1 && i2 < i1); // L(ixj, i)
                    bool doSwap = ((i & k) == 0) ? partnerFirst : !partnerFirst;
                    if (doSwap) {
                        ss[i] = s2; ss[ixj] = s1;
                        si[i] = i2; si[ixj] = i1;
                    }
                }
            }
            __syncthreads();
        }
    }

    // gather boxes in sorted order
    for (int i = tid; i < NB; i += 256) {
        int o = si[i];
        const float* b = boxes + ((size_t)o * N_CLS + c) * 4;
        float x1 = b[0], y1 = b[1], x2 = b[2], y2 = b[3];
        bx1[i] = x1; by1[i] = y1; bx2[i] = x2; by2[i] = y2;
        ar[i] = (x2 - x1) * (y2 - y1);
        supp[i] = 0;
    }
    __syncthreads();

    // greedy suppression
    for (int i = 0; i < NB - 1; ++i) {
        if (!supp[i]) {
            float X1 = bx1[i], Y1 = by1[i], X2 = bx2[i], Y2 = by2[i], A0 = ar[i];
            for (int jj = i + 1 + tid; jj < NB; jj += 256) {
                float ix1 = fmaxf(X1, bx1[jj]);
                float iy1 = fmaxf(Y1, by1[jj]);
                float ix2 = fminf(X2, bx2[jj]);
                float iy2 = fminf(Y2, by2[jj]);
                float inter = fmaxf(ix2 - ix1, 0.f) * fmaxf(iy2 - iy1, 0.f);
                float iou = inter / (A0 + ar[jj] - inter);
                if (iou > NMS_THRESH) supp[jj] = 1;
            }
        }
        __syncthreads();
    }

    // apply mask in place
    for (int i = tid; i < NB; i += 256) {
        if (supp[i]) att[(size_t)si[i] * 152 + c] = 0.f;
    }
}

// ---------------------------------------------------------------------------
// Host-side orchestration
// ---------------------------------------------------------------------------
extern "C" void kernel_launch(void* const* d_in, const int* in_sizes, int n_in,
                              void* d_out, int out_size, void* d_ws, size_t ws_size,
                              hipStream_t stream) {
    (void)in_sizes; (void)n_in; (void)out_size; (void)ws_size;
    const float* node_emb     = (const float*)d_in[0];   // 512 x 1024
    const float* edge_emb     = (const float*)d_in[1];   // 50000 x 1024
    const float* boxes        = (const float*)d_in[2];   // 512 x 151 x 4
    const float* nodes_schema = (const float*)d_in[3];   // 1024 x 151
    const float* edges_schema = (const float*)d_in[4];   // 1024 x 51

    float* out   = (float*)d_out;
    float* o_rec = out;                                   // raw_edge_class 50000*51
    float* o_he  = out + (size_t)E_EDGES * E_CLS;         // h_edge_emb     50000*51
    float* o_rnc = o_he + (size_t)E_EDGES * E_CLS;        // raw_node_class 512*151
    float* o_hne = o_rnc + (size_t)N_NODES * N_CLS;       // h_node_emb     512*1024

    // workspace layout (floats)
    float* ws   = (float*)d_ws;
    float* BpE  = ws;                    // edges_schema packed:  128 kpairs * 4 tiles  * 128 = 65536
    float* BpN  = BpE + 65536;           // nodes_schema packed:  128 kpairs * 10 tiles * 128 = 163840
    float* BpT  = BpN + 163840;          // nodes_schema^T packed: 19 kpairs * 64 tiles * 128 = 155648
    float* attw = BpT + 155648;          // att buffer 512 x 152

    // 1) pack B matrices into WMMA paired-fragment layout
    {
        int tE = 128 * 4 * 32;
        pack_b_kernel<<<(tE + 255) / 256, 256, 0, stream>>>(
            edges_schema, DIM, E_CLS, 1, E_CLS, 128, 4, BpE);
        int tN = 128 * 10 * 32;
        pack_b_kernel<<<(tN + 255) / 256, 256, 0, stream>>>(
            nodes_schema, DIM, N_CLS, 1, N_CLS, 128, 10, BpN);
        // B = nodes_schema^T (151 x 1024): B[k][n] = nodes_schema[n][k]
        int tT = 19 * 64 * 32;
        pack_b_kernel<<<(tT + 255) / 256, 256, 0, stream>>>(
            nodes_schema, N_CLS, 1, N_CLS, DIM, 19, 64, BpT);
    }

    // 2) raw_edge_class = edge_emb @ edges_schema   (memory-bound main kernel)
    {
        int mtiles = E_EDGES / 16;                 // 3125
        dim3 grid((mtiles + 7) / 8, 1);
        wmma_gemm_kernel<4><<<grid, 256, 0, stream>>>(
            edge_emb, DIM, BpE, 4, o_rec, E_CLS, mtiles, E_CLS, 128);
    }

    // 3) h_edge_emb = zeros
    {
        int n = E_EDGES * E_CLS;
        zero_kernel<<<(n + 255) / 256, 256, 0, stream>>>(o_he, n);
    }

    // 4) raw_node_class = node_emb @ nodes_schema
    {
        int mtiles = N_NODES / 16;                 // 32
        dim3 grid((mtiles + 7) / 8, 2);            // 10 N-tiles = 2 groups of 5
        wmma_gemm_kernel<5><<<grid, 256, 0, stream>>>(
            node_emb, DIM, BpN, 10, o_rnc, N_CLS, mtiles, N_CLS, 128);
    }

    // 5) att = softmax(raw_node_class); col 0 / pad col zeroed
    softmax_kernel<<<N_NODES / 8, 256, 0, stream>>>(o_rnc, attw);

    // 6) NMS per class, masks att in place
    nms_kernel<<<N_CLS - 1, 256, 0, stream>>>(boxes, attw);

    // 7) h_node_emb = att @ nodes_schema^T   (K = 151 padded to 152, 19 pairs)
    {
        int mtiles = N_NODES / 16;                 // 32
        dim3 grid((mtiles + 7) / 8, 16);           // 64 N-tiles = 16 groups of 4
        wmma_gemm_kernel<4><<<grid, 256, 0, stream>>>(
            attw, 152, BpT, 64, o_hne, DIM, mtiles, DIM, 19);
    }
}